// HyperGraphRep_49649821942022
// MI455X (gfx1250) — compile-verified
//
#include <hip/hip_runtime.h>
#include <hip/hip_bf16.h>

#define NN 100000   // nodes
#define EE 100000   // hyperedges
#define MM 1600000  // incidence pairs
#define DD 128      // embedding dim

typedef __attribute__((ext_vector_type(2))) float v2f;
typedef __attribute__((ext_vector_type(8))) float v8f;

// ---------------------------------------------------------------------------
// K0: zero the output (sv region + si-accumulator region) and the count arrays
// ---------------------------------------------------------------------------
__global__ __launch_bounds__(256) void hg_zero(float* __restrict__ out,
                                               float* __restrict__ ecnt,
                                               float* __restrict__ ncnt) {
    const long long t = (long long)blockIdx.x * 256 + threadIdx.x;
    const long long total = (long long)(NN + EE) * DD;  // 25.6M floats
    if (t < total) out[t] = 0.0f;
    if (t < EE) ecnt[t] = 0.0f;
    if (t < NN) ncnt[t] = 0.0f;
}

// ---------------------------------------------------------------------------
// K1: scatter node embeddings into hyperedge accumulators (si region of d_out)
// one wave (32 lanes) per incidence pair; each lane atomically adds 4 floats.
// Accumulators (51 MB) are L2-resident on the 192 MB L2.
// ---------------------------------------------------------------------------
__global__ __launch_bounds__(256) void hg_scatter_n2e(const int* __restrict__ hidx,
                                                      const float* __restrict__ node_embs,
                                                      float* __restrict__ eacc,
                                                      float* __restrict__ ecnt) {
    const long long gid = (long long)blockIdx.x * 256 + threadIdx.x;
    const int pair = (int)(gid >> 5);
    if (pair >= MM) return;
    const int lane = (int)(gid & 31);
    const int r = hidx[pair];        // node index
    const int c = hidx[MM + pair];   // hyperedge index
    const float4 v = *(const float4*)(node_embs + (long long)r * DD + lane * 4);
    float* dst = eacc + (long long)c * DD + lane * 4;
    atomicAdd(dst + 0, v.x);
    atomicAdd(dst + 1, v.y);
    atomicAdd(dst + 2, v.z);
    atomicAdd(dst + 3, v.w);
    if (lane == 0) atomicAdd(&ecnt[c], 1.0f);
}

// ---------------------------------------------------------------------------
// K2: per-hyperedge MLP with fp32 WMMA (V_WMMA_F32_16X16X4_F32).
// 16 edges per block; 8 waves, each wave owns one 16-column output tile.
// Reads the edge sums from `si`, overwrites `si` with the MLP output.
// K-loop split: 32 unconditional steps + 1 tail step (per-lane cndmask,
// no divergent branch in the hot loop).
// ---------------------------------------------------------------------------
__global__ __launch_bounds__(256) void hg_mlp(float* __restrict__ si,         // [E,128] in: sums, out: si
                                              const float* __restrict__ ecnt, // [E]
                                              const float* __restrict__ attr, // [E,3]
                                              const float* __restrict__ W1,   // [131,128]
                                              const float* __restrict__ b1,   // [128]
                                              const float* __restrict__ W2,   // [128,128]
                                              const float* __restrict__ b2) { // [128]
    __shared__ float As[16][132];   // [edge, 0..127 mean | 128..130 attr | 131 zero-pad]
    __shared__ float Hs[16][132];   // hidden activations (padded stride: conflict-free)
    __shared__ float inv[16];

    const int t  = threadIdx.x;
    const int e0 = blockIdx.x * 16;

    if (t < 16) {
        float c = ecnt[e0 + t];
        inv[t] = 1.0f / (c > 1.0f ? c : 1.0f);
    }
    __syncthreads();

    // load 16x128 edge sums -> means in LDS (512 float4 chunks, 2 per thread)
    for (int ch = t; ch < 512; ch += 256) {
        const int m  = ch >> 5;          // 0..15
        const int k4 = (ch & 31) << 2;   // 0..124
        const float4 s = *(const float4*)(si + (long long)(e0 + m) * DD + k4);
        const float sc = inv[m];
        As[m][k4 + 0] = s.x * sc;
        As[m][k4 + 1] = s.y * sc;
        As[m][k4 + 2] = s.z * sc;
        As[m][k4 + 3] = s.w * sc;
    }
    if (t < 64) {
        const int m = t >> 2, j = t & 3;
        As[m][128 + j] = (j < 3) ? attr[(e0 + m) * 3 + j] : 0.0f;
    }
    __syncthreads();

    const int lane = t & 31;
    const int w    = t >> 5;                 // wave id -> output tile
    const int m    = lane & 15;              // A-matrix row for this lane
    const int hs   = lane >> 4;              // half-select (K pair / M+8)
    const int n    = (w << 4) + (lane & 15); // output column 0..127

    const float bias1 = b1[n];               // hoisted: overlap with WMMA loop
    const float bias2 = b2[n];

    // ---- Layer 1 main: K = 0..127, no conditionals ----
    v8f acc = {};
    for (int kb = 0; kb < 32; ++kb) {
        const int k0 = (kb << 2) + (hs << 1);
        v2f a, b;
        a.x = As[m][k0];
        a.y = As[m][k0 + 1];
        b.x = W1[k0 * DD + n];
        b.y = W1[(k0 + 1) * DD + n];
        acc = __builtin_amdgcn_wmma_f32_16x16x4_f32(false, a, false, b,
                                                    (short)0, acc, false, false);
    }
    // ---- Layer 1 tail: K = 128..131 (rows 128..130 real, row 131 zero) ----
    {
        const int k0 = 128 + (hs << 1);      // 128 or 130
        v2f a, b;
        a.x = As[m][k0];                     // attr cols / zero pad in LDS
        a.y = As[m][k0 + 1];
        b.x = W1[k0 * DD + n];               // rows 128,130 both valid
        b.y = (hs == 0) ? W1[129 * DD + n] : 0.0f;  // lane-dependent -> cndmask
        acc = __builtin_amdgcn_wmma_f32_16x16x4_f32(false, a, false, b,
                                                    (short)0, acc, false, false);
    }
#pragma unroll
    for (int v = 0; v < 8; ++v) {
        const float x = acc[v] + bias1;
        acc[v] = x > 0.0f ? x : 0.0f;        // ReLU
    }
#pragma unroll
    for (int v = 0; v < 8; ++v)
        Hs[v + (hs << 3)][n] = acc[v];       // C layout: row = v + 8*halfsel
    __syncthreads();

    // ---- Layer 2: [16 x 128] @ [128 x 16] ----
    v8f acc2 = {};
    for (int kb = 0; kb < 32; ++kb) {
        const int k0 = (kb << 2) + (hs << 1);
        v2f a, b;
        a.x = Hs[m][k0];
        a.y = Hs[m][k0 + 1];
        b.x = W2[k0 * DD + n];
        b.y = W2[(k0 + 1) * DD + n];
        acc2 = __builtin_amdgcn_wmma_f32_16x16x4_f32(false, a, false, b,
                                                     (short)0, acc2, false, false);
    }
#pragma unroll
    for (int v = 0; v < 8; ++v)
        si[(long long)(e0 + v + (hs << 3)) * DD + n] = acc2[v] + bias2;
}

// ---------------------------------------------------------------------------
// K3: scatter hyperedge reps back into node accumulators (sv region of d_out)
// ---------------------------------------------------------------------------
__global__ __launch_bounds__(256) void hg_scatter_e2n(const int* __restrict__ hidx,
                                                      const float* __restrict__ si,
                                                      float* __restrict__ nacc,
                                                      float* __restrict__ ncnt) {
    const long long gid = (long long)blockIdx.x * 256 + threadIdx.x;
    const int pair = (int)(gid >> 5);
    if (pair >= MM) return;
    const int lane = (int)(gid & 31);
    const int r = hidx[pair];        // node index (destination)
    const int c = hidx[MM + pair];   // hyperedge index (source)
    const float4 v = *(const float4*)(si + (long long)c * DD + lane * 4);
    float* dst = nacc + (long long)r * DD + lane * 4;
    atomicAdd(dst + 0, v.x);
    atomicAdd(dst + 1, v.y);
    atomicAdd(dst + 2, v.z);
    atomicAdd(dst + 3, v.w);
    if (lane == 0) atomicAdd(&ncnt[r], 1.0f);
}

// ---------------------------------------------------------------------------
// K4: finalize sv = node_sum / max(count, 1) in place
// ---------------------------------------------------------------------------
__global__ __launch_bounds__(256) void hg_finalize(float* __restrict__ sv,
                                                   const float* __restrict__ ncnt) {
    const long long t = (long long)blockIdx.x * 256 + threadIdx.x;
    if (t >= (long long)NN * DD) return;
    const int node = (int)(t >> 7);
    float c = ncnt[node];
    c = c > 1.0f ? c : 1.0f;
    sv[t] = sv[t] / c;
}

// ---------------------------------------------------------------------------
extern "C" void kernel_launch(void* const* d_in, const int* in_sizes, int n_in,
                              void* d_out, int out_size, void* d_ws, size_t ws_size,
                              hipStream_t stream) {
    (void)in_sizes; (void)n_in; (void)out_size; (void)ws_size;

    const int*   hidx = (const int*)d_in[0];    // [2, M]
    const float* embs = (const float*)d_in[1];  // [N, 128]
    const float* attr = (const float*)d_in[2];  // [E, 3]
    const float* W1   = (const float*)d_in[3];  // [131, 128]
    const float* b1   = (const float*)d_in[4];  // [128]
    const float* W2   = (const float*)d_in[5];  // [128, 128]
    const float* b2   = (const float*)d_in[6];  // [128]

    float* out = (float*)d_out;
    float* sv  = out;                        // [N, 128]
    float* si  = out + (long long)NN * DD;   // [E, 128] (sums, then MLP output)

    float* ecnt = (float*)d_ws;              // [E]
    float* ncnt = ecnt + EE;                 // [N]

    // K0: zero accumulators (25.6M floats + counts)
    {
        const long long total = (long long)(NN + EE) * DD;
        const int blocks = (int)((total + 255) / 256);
        hg_zero<<<blocks, 256, 0, stream>>>(out, ecnt, ncnt);
    }
    // K1: scatter node embeddings -> edge sums (one wave per pair)
    {
        const long long threads = (long long)MM * 32;
        const int blocks = (int)((threads + 255) / 256);
        hg_scatter_n2e<<<blocks, 256, 0, stream>>>(hidx, embs, si, ecnt);
    }
    // K2: edge MLP with fp32 WMMA (16 edges / block)
    hg_mlp<<<EE / 16, 256, 0, stream>>>(si, ecnt, attr, W1, b1, W2, b2);
    // K3: scatter edge reps -> node sums
    {
        const long long threads = (long long)MM * 32;
        const int blocks = (int)((threads + 255) / 256);
        hg_scatter_e2n<<<blocks, 256, 0, stream>>>(hidx, si, sv, ncnt);
    }
    // K4: finalize node means
    {
        const long long total = (long long)NN * DD;
        const int blocks = (int)((total + 255) / 256);
        hg_finalize<<<blocks, 256, 0, stream>>>(sv, ncnt);
    }
}